// AFiReLoss_68513318305867
// MI455X (gfx1250) — compile-verified
//
#include <hip/hip_runtime.h>
#include <cstddef>

#define L_DIM 196
#define K_DIM 8192
#define B_DIM 64
#define LK ((size_t)L_DIM * K_DIM)        // 1,605,632
#define RECON_N 9633792                   // 64*3*224*224
#define RECON_N4 (RECON_N / 4)

typedef float v2f __attribute__((ext_vector_type(2)));
typedef float v8f __attribute__((ext_vector_type(8)));

// workspace offsets (in floats)
#define OFF_M    0
#define OFF_ROW  1605632
#define OFF_COL  1613824
#define OFF_CE   1614336
#define OFF_CNT  1616384
#define OFF_REC  1618432

// ---------------------------------------------------------------------------
// Kernel 1: M[l,k] = exp( proto[l,k] / 0.05 ), proto = EMA(prototype, mean_b T)
// Batch-mean over B=64 done with v_wmma_f32_16x16x4_f32: A = ones(16x4),
// B-matrix slots hold 4 distinct b-rows (permutation-invariant under ones-A).
// One wave produces 16 consecutive (l,k) outputs; 16 chained WMMAs cover b=0..63.
// ---------------------------------------------------------------------------
__global__ __launch_bounds__(256) void proto_exp_kernel(
    const float* __restrict__ tQ, const float* __restrict__ prot,
    const int* __restrict__ epoch, float* __restrict__ M) {
  const int wave = (blockIdx.x * 256 + threadIdx.x) >> 5;
  const int lane = threadIdx.x & 31;
  const int half = lane >> 4;
  const int col  = lane & 15;
  const size_t lk0 = (size_t)wave * 16;

  v8f acc = {0.f, 0.f, 0.f, 0.f, 0.f, 0.f, 0.f, 0.f};
  const v2f ones = {1.f, 1.f};
  const float* base = tQ + lk0 + col;

  #pragma unroll 4
  for (int s = 0; s < 16; ++s) {
    const int b0 = 4 * s;
    v2f bm;
    bm.x = base[(size_t)(b0 + half * 2)     * LK];
    bm.y = base[(size_t)(b0 + half * 2 + 1) * LK];
    acc = __builtin_amdgcn_wmma_f32_16x16x4_f32(false, ones, false, bm,
                                                (short)0, acc, false, false);
  }
  // acc[0] holds sum over b for column (lane&15), replicated across halves.
  const float mean = acc[0] * (1.f / 64.f);
  const float pv   = prot[lk0 + col];
  const float p    = (epoch[0] == 0) ? mean : (0.75f * pv + 0.25f * mean);
  const float e    = __expf(p * 20.f);   // exp(proto / 0.05)
  if (half == 0) M[lk0 + col] = e;
}

// ---------------------------------------------------------------------------
// Sinkhorn row sums: rsum[k] = sum_l M[l,k]  (contraction over L via ones-WMMA)
// One wave -> 16 k columns, 49 chunks of 4 l-rows (196 = 49*4).
// ---------------------------------------------------------------------------
__global__ __launch_bounds__(256) void rowsum_kernel(
    const float* __restrict__ M, float* __restrict__ rsum) {
  const int wave = (blockIdx.x * 256 + threadIdx.x) >> 5;
  const int lane = threadIdx.x & 31;
  const int half = lane >> 4;
  const int col  = lane & 15;
  const int k0   = wave * 16;

  v8f acc = {0.f, 0.f, 0.f, 0.f, 0.f, 0.f, 0.f, 0.f};
  const v2f ones = {1.f, 1.f};
  const float* base = M + k0 + col;

  for (int s = 0; s < 49; ++s) {
    const int l0 = 4 * s;
    v2f bm;
    bm.x = base[(size_t)(l0 + half * 2)     * K_DIM];
    bm.y = base[(size_t)(l0 + half * 2 + 1) * K_DIM];
    acc = __builtin_amdgcn_wmma_f32_16x16x4_f32(false, ones, false, bm,
                                                (short)0, acc, false, false);
  }
  if (half == 0) rsum[k0 + col] = acc[0];
}

// M[l,k] /= rsum[k] * 196   (float4 per thread; 401408 threads exactly)
__global__ __launch_bounds__(256) void rowscale_kernel(
    float* __restrict__ M, const float* __restrict__ rsum) {
  const int idx  = blockIdx.x * 256 + threadIdx.x;   // float4 index
  const int k    = (idx * 4) & (K_DIM - 1);
  float4 v = ((const float4*)M)[idx];
  float4 r = ((const float4*)rsum)[k >> 2];
  v.x /= (r.x * (float)L_DIM);
  v.y /= (r.y * (float)L_DIM);
  v.z /= (r.z * (float)L_DIM);
  v.w /= (r.w * (float)L_DIM);
  ((float4*)M)[idx] = v;
}

// csum[l] = sum_k M[l,k]  (contiguous; one block per l)
__global__ __launch_bounds__(256) void colsum_kernel(
    const float* __restrict__ M, float* __restrict__ csum) {
  __shared__ float red[256];
  const int tid = threadIdx.x;
  const float4* row = (const float4*)(M + (size_t)blockIdx.x * K_DIM);
  float s = 0.f;
  for (int i = tid; i < K_DIM / 4; i += 256) {
    const float4 v = row[i];
    s += (v.x + v.y) + (v.z + v.w);
  }
  red[tid] = s; __syncthreads();
  for (int st = 128; st > 0; st >>= 1) {
    if (tid < st) red[tid] += red[tid + st];
    __syncthreads();
  }
  if (tid == 0) csum[blockIdx.x] = red[0];
}

// M[l,k] /= csum[l] * factor  (factor = 8192 for iters 0,1; 1 for final iter,
// which folds the trailing *rows of the reference into the last normalization)
__global__ __launch_bounds__(256) void colscale_kernel(
    float* __restrict__ M, const float* __restrict__ csum, float factor) {
  const int idx = blockIdx.x * 256 + threadIdx.x;
  const int l   = (idx * 4) >> 13;                   // /8192
  const float c = csum[l] * factor;
  float4 v = ((const float4*)M)[idx];
  v.x /= c; v.y /= c; v.z /= c; v.w /= c;
  ((float4*)M)[idx] = v;
}

// ---------------------------------------------------------------------------
// wave32 butterfly + 8-slot LDS combine; deterministic fixed-order reduction.
// ---------------------------------------------------------------------------
__device__ inline float waveRed(float v, bool mx) {
  #pragma unroll
  for (int o = 16; o > 0; o >>= 1) {
    const float u = __shfl_xor(v, o, 32);
    v = mx ? fmaxf(v, u) : (v + u);
  }
  return v;
}

__device__ inline float blockRed(float v, bool mx, float* red8, int tid) {
  v = waveRed(v, mx);
  if ((tid & 31) == 0) red8[tid >> 5] = v;
  __syncthreads();
  float r = red8[0];
  #pragma unroll
  for (int i = 1; i < 8; ++i) r = mx ? fmaxf(r, red8[i]) : (r + red8[i]);
  __syncthreads();
  return r;
}

// ---------------------------------------------------------------------------
// CE kernel: block = (l, 8 b-rows). Teacher row staged in LDS (L2-resident
// source, 6.4MB total). Single b128 sweep per student row: register tile of
// 8 float4 per thread; fused max + dot(y,t); exp-sum from registers.
// per_patch = lse*sumT - sum_k y_k t_k  (y = 10x; dot needs no max shift).
// ---------------------------------------------------------------------------
__global__ __launch_bounds__(256) void ce_kernel(
    const float* __restrict__ sQ, const float* __restrict__ M,
    const int* __restrict__ labels, float* __restrict__ ce_part,
    float* __restrict__ cnt_part) {
  __shared__ float4 t_s[K_DIM / 4];
  __shared__ float red8[8];
  const int tid = threadIdx.x;
  const int l   = blockIdx.x;
  const int bc  = blockIdx.y;

  // stage teacher row (b128) + sumT
  const float4* trow = (const float4*)(M + (size_t)l * K_DIM);
  float st = 0.f;
  #pragma unroll
  for (int j = 0; j < 8; ++j) {
    const int i4 = tid + 256 * j;
    const float4 v = trow[i4];
    t_s[i4] = v;
    st += (v.x + v.y) + (v.z + v.w);
  }
  __syncthreads();
  const float sumT = blockRed(st, false, red8, tid);

  float csum = 0.f, cntv = 0.f;
  for (int bi = 0; bi < 8; ++bi) {
    const int b = bc * 8 + bi;
    const float4* row = (const float4*)(sQ + ((size_t)b * L_DIM + l) * K_DIM);
    if (bi < 7) {  // prefetch next student row (32KB: 256 thr * 128B)
      const float* nxt = sQ + ((size_t)(b + 1) * L_DIM + l) * K_DIM;
      __builtin_prefetch(nxt + tid * 32, 0, 0);
    }
    float4 x[8];
    float mx = -3.4e38f, dot = 0.f;
    #pragma unroll
    for (int j = 0; j < 8; ++j) {
      const int i4 = tid + 256 * j;
      const float4 v = row[i4];
      const float4 t = t_s[i4];
      x[j] = v;
      mx  = fmaxf(mx, fmaxf(fmaxf(v.x, v.y), fmaxf(v.z, v.w)));
      dot = fmaf(t.x, v.x, dot);
      dot = fmaf(t.y, v.y, dot);
      dot = fmaf(t.z, v.z, dot);
      dot = fmaf(t.w, v.w, dot);
    }
    const float m = blockRed(mx * 10.f, true, red8, tid);

    float se = 0.f;
    #pragma unroll
    for (int j = 0; j < 8; ++j) {
      se += __expf(x[j].x * 10.f - m);
      se += __expf(x[j].y * 10.f - m);
      se += __expf(x[j].z * 10.f - m);
      se += __expf(x[j].w * 10.f - m);
    }
    const float S    = blockRed(se, false, red8, tid);
    const float dotY = blockRed(dot * 10.f, false, red8, tid);

    if (tid == 0) {
      const float lse = m + logf(S);
      const float pp  = lse * sumT - dotY;
      if (labels[b * L_DIM + l] == 0) { csum += pp; cntv += 1.f; }
    }
  }
  if (tid == 0) {
    const int id = bc * gridDim.x + l;
    ce_part[id]  = csum;
    cnt_part[id] = cntv;
  }
}

// ---------------------------------------------------------------------------
__global__ __launch_bounds__(256) void recon_kernel(
    const float* __restrict__ r, const float* __restrict__ lb,
    float* __restrict__ part) {
  __shared__ float red[256];
  const int tid = threadIdx.x;
  float s = 0.f;
  for (int i = blockIdx.x * 256 + tid; i < RECON_N4; i += gridDim.x * 256) {
    const float4 a = ((const float4*)r)[i];
    const float4 b = ((const float4*)lb)[i];
    const float dx = a.x - b.x, dy = a.y - b.y, dz = a.z - b.z, dw = a.w - b.w;
    s = fmaf(dx, dx, s); s = fmaf(dy, dy, s);
    s = fmaf(dz, dz, s); s = fmaf(dw, dw, s);
  }
  red[tid] = s; __syncthreads();
  for (int st = 128; st > 0; st >>= 1) {
    if (tid < st) red[tid] += red[tid + st];
    __syncthreads();
  }
  if (tid == 0) part[blockIdx.x] = red[0];
}

__global__ __launch_bounds__(256) void finalize_kernel(
    const float* __restrict__ ce, const float* __restrict__ cnt,
    const float* __restrict__ rec, int nce, int nrec, float* __restrict__ out) {
  __shared__ float r1[256], r2[256], r3[256];
  const int tid = threadIdx.x;
  float a = 0.f, b = 0.f, c = 0.f;
  for (int i = tid; i < nce; i += 256) { a += ce[i]; b += cnt[i]; }
  for (int i = tid; i < nrec; i += 256) c += rec[i];
  r1[tid] = a; r2[tid] = b; r3[tid] = c; __syncthreads();
  for (int s = 128; s > 0; s >>= 1) {
    if (tid < s) { r1[tid] += r1[tid+s]; r2[tid] += r2[tid+s]; r3[tid] += r3[tid+s]; }
    __syncthreads();
  }
  if (tid == 0) out[0] = r1[0] / r2[0] + r3[0] * (1.f / (float)RECON_N);
}

// ---------------------------------------------------------------------------
extern "C" void kernel_launch(void* const* d_in, const int* in_sizes, int n_in,
                              void* d_out, int out_size, void* d_ws, size_t ws_size,
                              hipStream_t stream) {
  (void)in_sizes; (void)n_in; (void)out_size; (void)ws_size;
  const float* sQ    = (const float*)d_in[0];
  const float* tQ    = (const float*)d_in[1];
  const float* recon = (const float*)d_in[2];
  const float* label = (const float*)d_in[3];
  const float* prot  = (const float*)d_in[4];
  const int*   plab  = (const int*)d_in[5];
  const int*   epoch = (const int*)d_in[6];
  float* out = (float*)d_out;
  float* ws  = (float*)d_ws;

  float* M    = ws + OFF_M;
  float* rsum = ws + OFF_ROW;
  float* csum = ws + OFF_COL;
  float* ceP  = ws + OFF_CE;
  float* cntP = ws + OFF_CNT;
  float* recP = ws + OFF_REC;

  // 1) proto update + exp  (1,605,632 outputs / 128 per block)
  proto_exp_kernel<<<12544, 256, 0, stream>>>(tQ, prot, epoch, M);

  // 2) Sinkhorn: 3 iterations; initial global normalization cancels exactly
  for (int it = 0; it < 3; ++it) {
    rowsum_kernel<<<64, 256, 0, stream>>>(M, rsum);
    rowscale_kernel<<<1568, 256, 0, stream>>>(M, rsum);
    colsum_kernel<<<L_DIM, 256, 0, stream>>>(M, csum);
    const float factor = (it == 2) ? 1.f : (float)K_DIM;  // fold *rows into last
    colscale_kernel<<<1568, 256, 0, stream>>>(M, csum, factor);
  }

  // 3) masked cross-entropy (grid = 196 l-rows x 8 b-chunks)
  ce_kernel<<<dim3(L_DIM, 8), 256, 0, stream>>>(sQ, M, plab, ceP, cntP);

  // 4) reconstruction MSE partials
  recon_kernel<<<1024, 256, 0, stream>>>(recon, label, recP);

  // 5) combine
  finalize_kernel<<<1, 256, 0, stream>>>(ceP, cntP, recP, 1568, 1024, out);
}